// gcn_67834713473661
// MI455X (gfx1250) — compile-verified
//
#include <hip/hip_runtime.h>

#define NN 64
#define CC 64
#define TT 256
#define VV 25
#define SS 3
#define OCC 64
#define ICC 32
#define TV (TT*VV)                          // 6400
#define A_ELEMS ((size_t)NN*SS*ICC*TV)      // 39,321,600 floats per a/b buffer
#define ATT_STRIDE 1024                     // padded 32x32 per (n,s)

typedef float v2f __attribute__((ext_vector_type(2)));
typedef float v8f __attribute__((ext_vector_type(8)));

#define V8F_ZERO {0.f,0.f,0.f,0.f,0.f,0.f,0.f,0.f}

__device__ __forceinline__ v8f wmma_f32(v2f a, v2f b, v8f c) {
  // D = A(16x4) * B(4x16) + C, fp32 exact
  return __builtin_amdgcn_wmma_f32_16x16x4_f32(false, a, false, b, (short)0, c, false, false);
}

// ---------------------------------------------------------------------------
// Kernel 1: a = Wa[s]@x[n]+ba, b = Wb[s]@x[n]+bb.
// One wave per 16-wide column strip, computing all 4 stacked M-tiles (a rows
// 0..31, b rows 0..31) so the x tile (B operand) is loaded once per k-step.
// a stored [ns][i][t][v], b likewise.
// ---------------------------------------------------------------------------
__global__ void k_conv_ab(const float* __restrict__ x,
                          const float* __restrict__ Wa, const float* __restrict__ ba,
                          const float* __restrict__ Wb, const float* __restrict__ bb,
                          float* __restrict__ wsa, float* __restrict__ wsb) {
  int wid  = blockIdx.x * (blockDim.x >> 5) + (threadIdx.x >> 5);
  int lane = threadIdx.x & 31;
  int l16 = lane & 15, half = lane >> 4;

  int nt = wid % 400;            // N tile (tv)
  int ns = wid / 400;            // n*S + s
  int s  = ns % SS;
  int n  = ns / SS;

  int col = nt * 16 + l16;       // tv column 0..6399
  const float* xb = x + (size_t)n * CC * TV;
  const float* W0 = Wa + (size_t)(s * ICC + l16)      * CC;  // a rows 0..15
  const float* W1 = Wa + (size_t)(s * ICC + 16 + l16) * CC;  // a rows 16..31
  const float* W2 = Wb + (size_t)(s * ICC + l16)      * CC;  // b rows 0..15
  const float* W3 = Wb + (size_t)(s * ICC + 16 + l16) * CC;  // b rows 16..31

  v8f acc0 = V8F_ZERO, acc1 = V8F_ZERO, acc2 = V8F_ZERO, acc3 = V8F_ZERO;
  for (int kk = 0; kk < CC; kk += 4) {
    int k0 = kk + half * 2;
    v2f bv, a0, a1, a2, a3;
#pragma unroll
    for (int j = 0; j < 2; ++j) {
      bv[j] = xb[(size_t)(k0 + j) * TV + col];
      a0[j] = W0[k0 + j];
      a1[j] = W1[k0 + j];
      a2[j] = W2[k0 + j];
      a3[j] = W3[k0 + j];
    }
    acc0 = wmma_f32(a0, bv, acc0);
    acc1 = wmma_f32(a1, bv, acc1);
    acc2 = wmma_f32(a2, bv, acc2);
    acc3 = wmma_f32(a3, bv, acc3);
  }
  size_t base = (size_t)ns * ICC * TV;
#pragma unroll
  for (int j = 0; j < 8; ++j) {
    int r = j + half * 8;
    wsa[base + (size_t)(r)      * TV + col] = acc0[j] + ba[s * ICC + r];
    wsa[base + (size_t)(r + 16) * TV + col] = acc1[j] + ba[s * ICC + r + 16];
    wsb[base + (size_t)(r)      * TV + col] = acc2[j] + bb[s * ICC + r];
    wsb[base + (size_t)(r + 16) * TV + col] = acc3[j] + bb[s * ICC + r + 16];
  }
}

// ---------------------------------------------------------------------------
// Kernel 2: scores[v][w] = (1/8192) * sum_k a[k][v]*b[k][w]; softmax over v;
// + A[s]; write padded 32x32 att. One block per (n,s), 4 waves, one per 16x16 tile.
// ---------------------------------------------------------------------------
__global__ void k_scores(const float* __restrict__ wsa, const float* __restrict__ wsb,
                         const float* __restrict__ Amat, float* __restrict__ wsatt) {
  __shared__ float sc[2][32][33];
  int ns = blockIdx.x;
  int s  = ns % SS;
  int wave = threadIdx.x >> 5;
  int lane = threadIdx.x & 31;
  int l16 = lane & 15, half = lane >> 4;
  int mt = (wave >> 1) & 1;
  int nt = wave & 1;
  int kh = wave >> 2;                       // K half
  size_t abase = (size_t)ns * ICC * TV;     // a/b are [k=i*T+t][v] contiguous
  int vrow = mt * 16 + l16;
  int wcol = nt * 16 + l16;
  v8f acc = V8F_ZERO;
  int kbeg = kh * (ICC * TT / 2);
  int kend = kbeg + (ICC * TT / 2);
  for (int kk = kbeg; kk < kend; kk += 4) {
    int k0 = kk + half * 2;
    v2f av, bv;
#pragma unroll
    for (int j = 0; j < 2; ++j) {
      av[j] = wsa[abase + (size_t)(k0 + j) * VV + vrow];
      bv[j] = wsb[abase + (size_t)(k0 + j) * VV + wcol];
    }
    acc = wmma_f32(av, bv, acc);
  }
#pragma unroll
  for (int j = 0; j < 8; ++j)
    sc[kh][mt * 16 + j + half * 8][wcol] = acc[j];
  __syncthreads();
  if (threadIdx.x < 32) {
    int w = threadIdx.x;                    // one column per lane
    const float inv = 1.0f / (float)(ICC * TT);
    float mx = -3.4e38f;
    for (int v = 0; v < VV; ++v)
      mx = fmaxf(mx, (sc[0][v][w] + sc[1][v][w]) * inv);
    float e[VV];
    float sum = 0.f;
    for (int v = 0; v < VV; ++v) {
      float t = (sc[0][v][w] + sc[1][v][w]) * inv;
      float ev = __expf(t - mx);
      e[v] = ev; sum += ev;
    }
    float r = 1.0f / sum;
    size_t ob = (size_t)ns * ATT_STRIDE;
    for (int v = 0; v < 32; ++v) {
      float o = 0.f;
      if (v < VV && w < VV)
        o = e[v] * r + Amat[((size_t)s * VV + v) * VV + w];
      wsatt[ob + v * 32 + w] = o;           // zero-padded 32x32
    }
  }
}

// ---------------------------------------------------------------------------
// Kernel 3: xw = Wd[s] @ x[n]  (M=64, K=64, N=6400); overwrites a/b region.
// N columns enumerated in permuted order colp = v*256 + t so the output is
// stored TRANSPOSED as [ns][o][v][t] -> kernel 4's A-operand loads become
// lane-contiguous. Stores stay coalesced; x reads are strided but L2-resident.
// ---------------------------------------------------------------------------
__global__ void k_conv_d(const float* __restrict__ x, const float* __restrict__ Wd,
                         float* __restrict__ wsxw) {
  int wid  = blockIdx.x * (blockDim.x >> 5) + (threadIdx.x >> 5);
  int lane = threadIdx.x & 31;
  int l16 = lane & 15, half = lane >> 4;
  int nt = wid % 400;
  int ns = wid / 400;
  int s  = ns % SS, n = ns / SS;
  int colp = nt * 16 + l16;                 // permuted column: v*256 + t
  int vcol = colp >> 8;                     // uniform per 16-lane group
  int tcol = colp & 255;
  const float* xb = x + (size_t)n * CC * TV + (size_t)tcol * VV + vcol; // + k*TV
  const float* W0 = Wd + (size_t)(s * OCC + l16) * CC;
  const float* W1 = W0 + 16 * CC;
  const float* W2 = W0 + 32 * CC;
  const float* W3 = W0 + 48 * CC;

  v8f acc0 = V8F_ZERO, acc1 = V8F_ZERO, acc2 = V8F_ZERO, acc3 = V8F_ZERO;
  for (int kk = 0; kk < CC; kk += 4) {
    int k0 = kk + half * 2;
    v2f bv, a0, a1, a2, a3;
#pragma unroll
    for (int j = 0; j < 2; ++j) {
      bv[j] = xb[(size_t)(k0 + j) * TV];
      a0[j] = W0[k0 + j];
      a1[j] = W1[k0 + j];
      a2[j] = W2[k0 + j];
      a3[j] = W3[k0 + j];
    }
    acc0 = wmma_f32(a0, bv, acc0);
    acc1 = wmma_f32(a1, bv, acc1);
    acc2 = wmma_f32(a2, bv, acc2);
    acc3 = wmma_f32(a3, bv, acc3);
  }
  // layout [ns][o][v][t]: addr = base + o*6400 + (v*256 + t) = base + o*TV + colp
  size_t base = (size_t)ns * OCC * TV;
#pragma unroll
  for (int j = 0; j < 8; ++j) {
    int r = j + half * 8;
    wsxw[base + (size_t)(r)      * TV + colp] = acc0[j];
    wsxw[base + (size_t)(r + 16) * TV + colp] = acc1[j];
    wsxw[base + (size_t)(r + 32) * TV + colp] = acc2[j];
    wsxw[base + (size_t)(r + 48) * TV + colp] = acc3[j];
  }
}

// ---------------------------------------------------------------------------
// Kernel 4: y[n,o,t,w] = sum_s sum_v xwT[n,s,o,v,t]*att[n,s,v,w] + bias + BN.
// One wave per M-tile of flattened (o,t); computes BOTH w tiles so each
// (coalesced) xw A-load is reused twice. K = 32 (padded V), predicate v<25.
// ---------------------------------------------------------------------------
__global__ void k_out(const float* __restrict__ wsxw, const float* __restrict__ wsatt,
                      const float* __restrict__ bd, const float* __restrict__ gamma,
                      const float* __restrict__ beta, float* __restrict__ out) {
  int wid  = blockIdx.x * (blockDim.x >> 5) + (threadIdx.x >> 5);
  int lane = threadIdx.x & 31;
  int l16 = lane & 15, half = lane >> 4;
  int mt = wid & 1023;
  int n  = wid >> 10;
  int o  = mt >> 4;                         // uniform over tile
  int t  = (mt * 16 + l16) & 255;           // lane-contiguous

  v8f acc0 = V8F_ZERO, acc1 = V8F_ZERO;
  for (int s = 0; s < SS; ++s) {
    int ns = n * SS + s;
    const float* xwb = wsxw + (size_t)(ns * OCC + o) * TV + t;   // + v*256
    const float* atb = wsatt + (size_t)ns * ATT_STRIDE;          // + v*32 + w
#pragma unroll
    for (int kk = 0; kk < 32; kk += 4) {
      int k0 = kk + half * 2;
      v2f av, b0, b1;
#pragma unroll
      for (int j = 0; j < 2; ++j) {
        int v = k0 + j;
        av[j] = (v < VV) ? xwb[(size_t)v * 256] : 0.0f;  // K padding predicate
        b0[j] = atb[v * 32 + l16];
        b1[j] = atb[v * 32 + 16 + l16];
      }
      acc0 = wmma_f32(av, b0, acc0);
      acc1 = wmma_f32(av, b1, acc1);
    }
  }
  float bds = bd[o] + bd[OCC + o] + bd[2 * OCC + o];
  float g   = gamma[o] * rsqrtf(1.0f + 1e-5f);
  float bt  = beta[o];
#pragma unroll
  for (int j = 0; j < 8; ++j) {
    int mr = mt * 16 + j + half * 8;
    int tt = mr & 255;
    size_t ob = (((size_t)n * OCC + o) * TT + tt) * VV;
    out[ob + l16] = (acc0[j] + bds) * g + bt;            // w = l16 < 16 always valid
    if (l16 < VV - 16)                                   // w = 16+l16 < 25
      out[ob + 16 + l16] = (acc1[j] + bds) * g + bt;
  }
}

extern "C" void kernel_launch(void* const* d_in, const int* in_sizes, int n_in,
                              void* d_out, int out_size, void* d_ws, size_t ws_size,
                              hipStream_t stream) {
  (void)in_sizes; (void)n_in; (void)out_size; (void)ws_size;
  const float* x  = (const float*)d_in[0];
  const float* A  = (const float*)d_in[1];
  const float* Wa = (const float*)d_in[2];
  const float* ba = (const float*)d_in[3];
  const float* Wb = (const float*)d_in[4];
  const float* bb = (const float*)d_in[5];
  const float* Wd = (const float*)d_in[6];
  const float* bd = (const float*)d_in[7];
  const float* gm = (const float*)d_in[8];
  const float* bt = (const float*)d_in[9];
  float* out = (float*)d_out;

  float* wsf   = (float*)d_ws;
  float* wsa   = wsf;                 // [192][32][6400]
  float* wsb   = wsf + A_ELEMS;       // [192][32][6400]
  float* wsxw  = wsf;                 // [192][64][25][256] reuses a+b after k_scores
  float* wsatt = wsf + 2 * A_ELEMS;   // [192][32][32] padded attention

  // 192 (n,s) * 400 column strips = 76800 waves / 8 per block
  k_conv_ab<<<9600, dim3(256), 0, stream>>>(x, Wa, ba, Wb, bb, wsa, wsb);
  // one block per (n,s)
  k_scores<<<NN * SS, dim3(128), 0, stream>>>(wsa, wsb, A, wsatt);
  k_conv_d<<<9600, dim3(256), 0, stream>>>(x, Wd, wsxw);
  // 64*1024 M-tiles (each wave does both w tiles) = 65536 waves / 8 per block
  k_out<<<8192, dim3(256), 0, stream>>>(wsxw, wsatt, bd, gm, bt, out);
}